// TopKGating_71098888618611
// MI455X (gfx1250) — compile-verified
//
#include <hip/hip_runtime.h>
#include <math.h>

typedef __attribute__((ext_vector_type(2))) float v2f;
typedef __attribute__((ext_vector_type(8))) float v8f;

#define TPB    128          // tokens per block
#define KC     32           // K chunk (floats); 1024/KC = 32 chunks
#define XS     36           // padded x-tile row stride (covers all 64 LDS banks)
#define WSTR   36           // padded W row stride
#define LSTR   9            // padded logits row stride
#define XBUF   (TPB * XS)   // 4608 floats per x buffer
#define WBUF   (16 * WSTR)  // 576 floats per W buffer (rows 8..15 stay zero)

// CDNA5 async global->LDS copy (ASYNCcnt-tracked, no VGPR round trip).
__device__ __forceinline__ void async_copy_b128(unsigned lds_off, const void* gptr) {
    asm volatile("global_load_async_to_lds_b128 %0, %1, off"
                 :: "v"(lds_off), "v"(gptr)
                 : "memory");
}
// Every wave issues exactly 5 async instructions per stage; loads retire in
// order, so waiting down to 5 guarantees the previous stage fully landed.
__device__ __forceinline__ void wait_async_prev() {
    asm volatile("s_wait_asynccnt 0x5" ::: "memory");
}
__device__ __forceinline__ void wait_async0() {
    asm volatile("s_wait_asynccnt 0x0" ::: "memory");
}
__device__ __forceinline__ unsigned lds_off_of(const void* p) {
    // Flat addr of __shared__: aperture in [63:32], LDS byte offset in [31:0].
    return (unsigned)(uintptr_t)p;
}

__global__ __launch_bounds__(256) void topk_gate_wmma(
    const float* __restrict__ x, const float* __restrict__ W,
    const float* __restrict__ b, int* __restrict__ out_idx,
    float* __restrict__ out_val, int n_tokens)
{
    __shared__ float arena[3 * XBUF + 3 * WBUF];   // 60.8 KB
    float* lds_w = arena + 3 * XBUF;
    float* lds_l = arena;                          // aliases x buf 0 (used after GEMM)

    const int tid  = threadIdx.x;
    const int wv   = tid >> 5;        // wave id 0..7 -> 16-token tile
    const int lane = tid & 31;
    const int half = lane >> 4;       // selects K pair within WMMA fragment
    const int l16  = lane & 15;       // M (for A) / N (for B) index
    const int row0 = blockIdx.x * TPB;

    // ---- async stage of one K chunk into buffer set `buf` (0..2) ----
    // Exactly 5 async instructions per wave (4 x-tile + 1 W), uniform.
    auto stage = [&](int buf, int kc) {
        float* xb = arena + buf * XBUF;
#pragma unroll
        for (int i = 0; i < 4; ++i) {             // 128 rows x 32 cols, 4 float4/thread
            int idx = tid + i * 256;              // 0..1023
            int r   = idx >> 3;                   // 8 float4 per row
            int c4  = (idx & 7) << 2;
            int gr  = row0 + r;
            if (gr >= n_tokens) gr = n_tokens - 1;
            async_copy_b128(lds_off_of(&xb[r * XS + c4]),
                            x + (size_t)gr * 1024 + kc + c4);
        }
        // W rows 0..7: 64 float4 slots spread over all 8 waves (8 lanes each)
        if ((tid & 3) == 0) {
            int s  = tid >> 2;                    // 0..63
            int e  = s >> 3;
            int c4 = (s & 7) << 2;
            async_copy_b128(lds_off_of(&lds_w[buf * WBUF + e * WSTR + c4]),
                            W + (size_t)e * 1024 + kc + c4);
        }
    };

    // ---- zero W pad rows (8..15) of all 3 buffers, once ----
    for (int idx = tid; idx < 3 * 8 * WSTR; idx += 256) {
        int buf = idx / (8 * WSTR);
        int off = idx % (8 * WSTR);
        lds_w[buf * WBUF + 8 * WSTR + off] = 0.0f;
    }

    v8f acc = {};   // 16x16 f32 accumulator (cols 0..7 valid)

    // ---- prologue: fill stages 0 and 1 ----
    stage(0, 0);
    stage(1, KC);
    wait_async_prev();      // stage 0 landed (stage 1 may be in flight)
    __syncthreads();

    int cur = 0;
    for (int kc = 0; kc < 1024; kc += KC) {
        const int nx = kc + 2 * KC;
        const int buf_next2 = (cur + 2 >= 3) ? cur - 1 : cur + 2;
        if (nx < 1024) stage(buf_next2, nx);   // issue stage k+2

        const float* xb = arena + cur * XBUF;
        const float* wb = lds_w + cur * WBUF;
        // ---- 8 WMMA steps over current chunk ----
#pragma unroll
        for (int kk = 0; kk < KC; kk += 4) {
            const int kcol = kk + 2 * half;
            v2f a  = *(const v2f*)(&xb[(wv * 16 + l16) * XS + kcol]);  // A: row=l16
            v2f bm = *(const v2f*)(&wb[l16 * WSTR + kcol]);            // B: col=l16 (padded)
            acc = __builtin_amdgcn_wmma_f32_16x16x4_f32(
                false, a, false, bm, (short)0, acc, false, false);
        }

        // ensure stage k+1 landed before everyone swaps to it
        if (nx < 1024)            wait_async_prev();
        else if (kc + KC < 1024)  wait_async0();
        __syncthreads();
        cur = (cur + 1 >= 3) ? 0 : cur + 1;
    }

    // ---- scatter logits: acc[i] -> row (wv*16 + half*8 + i), col l16 ----
    if (l16 < 8) {
#pragma unroll
        for (int i = 0; i < 8; ++i) {
            lds_l[(wv * 16 + half * 8 + i) * LSTR + l16] = acc[i];
        }
    }
    __syncthreads();

    // ---- per-token softmax + top-2 (one thread per token) ----
    if (tid < TPB) {
        const int row = row0 + tid;
        if (row < n_tokens) {
            float l[8];
#pragma unroll
            for (int e = 0; e < 8; ++e) l[e] = lds_l[tid * LSTR + e] + b[e];

            int   i1 = 0;    float v1 = l[0];          // lowest index wins ties
#pragma unroll
            for (int e = 1; e < 8; ++e) { if (l[e] > v1) { v1 = l[e]; i1 = e; } }
            int   i2 = -1;   float v2 = -INFINITY;
#pragma unroll
            for (int e = 0; e < 8; ++e) {
                if (e != i1 && l[e] > v2) { v2 = l[e]; i2 = e; }
            }
            float s = 0.0f;
#pragma unroll
            for (int e = 0; e < 8; ++e) s += __expf(l[e] - v1);
            float inv = 1.0f / s;

            int2 iv; iv.x = i1; iv.y = i2;
            *(int2*)&out_idx[(size_t)row * 2] = iv;
            float2 fv; fv.x = inv; fv.y = __expf(v2 - v1) * inv;
            *(float2*)&out_val[(size_t)row * 2] = fv;
        }
    }
}

extern "C" void kernel_launch(void* const* d_in, const int* in_sizes, int n_in,
                              void* d_out, int out_size, void* d_ws, size_t ws_size,
                              hipStream_t stream) {
    const float* x = (const float*)d_in[0];
    const float* W = (const float*)d_in[1];
    const float* b = (const float*)d_in[2];
    const int n = in_sizes[0] / 1024;            // NUM_TOKENS

    // d_out layout (tuple concat): [n*2] int32 indices, then [n*2] f32 values
    int*   out_idx = (int*)d_out;
    float* out_val = (float*)d_out + (size_t)2 * n;

    dim3 grid((n + TPB - 1) / TPB);
    dim3 block(256);
    hipLaunchKernelGGL(topk_gate_wmma, grid, block, 0, stream,
                       x, W, b, out_idx, out_val, n);
}